// WrapSoftmax_19971597927200
// MI455X (gfx1250) — compile-verified
//
#include <hip/hip_runtime.h>
#include <hip/hip_bf16.h>

typedef float v2f __attribute__((ext_vector_type(2)));
typedef float v8f __attribute__((ext_vector_type(8)));
typedef unsigned u32x4 __attribute__((ext_vector_type(4)));
typedef int i32x4 __attribute__((ext_vector_type(4)));
typedef int i32x8 __attribute__((ext_vector_type(8)));

#define NUM_CLASSES 100
#define NQ      256      // queries
#define DCODE   64       // hash code width
#define NDB     100000   // database rows
#define NTILES  6250     // NDB / 16
#define KDIM    2048     // input feature dim
#define HSTRIDE 2048     // histogram row stride (max bins)
#define NB1     1024     // level-1 bins (10 bits)
#define NB23    2048     // level-2/3 bins (11 bits)
#define WPB     8        // waves per block for histogram kernels
#define WPBC    4        // waves per block for TDM count kernel
#define NBX     40       // n-tile blocks
#define MAX_CAND 16384
#define ROWP    68       // staged row stride in DWORDs (64 + 4 TDM pad)

#if __has_builtin(__builtin_amdgcn_tensor_load_to_lds)
#define HAVE_TDM 1
#else
#define HAVE_TDM 0
#endif

// Monotonic float -> uint key: larger float => larger key.
__device__ __forceinline__ unsigned orderKey(float f) {
  unsigned u = __float_as_uint(f);
  return (u & 0x80000000u) ? ~u : (u | 0x80000000u);
}

// 16(queries) x 16(db rows) dot tile over K=64: 16 chained v_wmma_f32_16x16x4_f32.
// Per ISA 7.12.2: lane<16 holds K={0,1}, lane>=16 K={2,3}; B fragment is a
// contiguous float2 of the (transposed-use) codes row.
__device__ __forceinline__ v8f dot_tile(const float* __restrict__ outm,
                                        const float* __restrict__ codes,
                                        int q0, int n0, int lane) {
  const int half = lane >> 4;
  const int l15  = lane & 15;
  v8f acc = {};
  const float* arow = outm  + (size_t)(q0 + l15) * DCODE + 2 * half;
  const float* brow = codes + (size_t)(n0 + l15) * DCODE + 2 * half;
#pragma unroll
  for (int kk = 0; kk < 16; ++kk) {
    v2f a = *(const v2f*)(arow + 4 * kk);
    v2f b = *(const v2f*)(brow + 4 * kk);
    acc = __builtin_amdgcn_wmma_f32_16x16x4_f32(false, a, false, b,
                                                (short)0, acc, false, false);
  }
  return acc;
}

// Same, but B fragments come from an LDS-staged tile (row stride ROWP dwords).
__device__ __forceinline__ v8f dot_tile_lds(const float* __restrict__ outm,
                                            const float* brow,  // LDS, pre-offset
                                            int q0, int lane) {
  const int half = lane >> 4;
  const int l15  = lane & 15;
  v8f acc = {};
  const float* arow = outm + (size_t)(q0 + l15) * DCODE + 2 * half;
#pragma unroll
  for (int kk = 0; kk < 16; ++kk) {
    v2f a = *(const v2f*)(arow + 4 * kk);
    v2f b = *(const v2f*)(brow + 4 * kk);
    acc = __builtin_amdgcn_wmma_f32_16x16x4_f32(false, a, false, b,
                                                (short)0, acc, false, false);
  }
  return acc;
}

#if HAVE_TDM
// Issue a TDM load of a 16-row x 64-dword codes tile into LDS at ldsOff.
// D# per CDNA5 ISA: 2D tensor, data_size=4B,
// tensor_dim0=64, tensor_dim0_stride=64, tile 64x16, pad 4 dwords after every
// 64 dwords (row stride in LDS becomes ROWP=68 -> bank-conflict-free reads).
__device__ __forceinline__ void tdm_load_tile(unsigned ldsOff, const float* gsrc) {
  unsigned long long ga = (unsigned long long)(uintptr_t)gsrc;
  u32x4 g0;
  g0[0] = 1u;                                      // count=1, user descriptor
  g0[1] = ldsOff;                                  // lds_addr
  g0[2] = (unsigned)(ga & 0xFFFFFFFFu);            // global_addr[31:0]
  g0[3] = (unsigned)((ga >> 32) & 0x1FFFFFFu)      // global_addr[56:32]
        | (2u << 30);                              // type=2 ("image")
  i32x8 g1;
  g1[0] = (int)((2u << 16)                         // data_size = 4 bytes
        | (1u << 20)                               // pad_enable
        | (5u << 22)                               // pad_interval: every 64 dw
        | (3u << 25));                             // pad_amount: 4 dw
  g1[1] = (int)(64u << 16);                        // tensor_dim0 = 64 (low 16)
  g1[2] = (int)(0x86A0u << 16);                    // tensor_dim1 low16 (100000)
  g1[3] = (int)((64u << 16) | 1u);                 // tile_dim0=64 | dim1 high16
  g1[4] = 16;                                      // tile_dim1 = 16
  g1[5] = 64;                                      // tensor_dim0_stride = 64
  g1[6] = 0;
  g1[7] = 0;
  i32x4 g2 = {}; i32x4 g3 = {};
#if defined(__clang_major__) && (__clang_major__ >= 23)
  i32x8 g4 = {};
  __builtin_amdgcn_tensor_load_to_lds(g0, g1, g2, g3, g4, 0);
#else
  __builtin_amdgcn_tensor_load_to_lds(g0, g1, g2, g3, 0);
#endif
}
#endif

// ---- Kernel 1: out[256,64] = x[256,2048] @ W[2048,64] (WMMA fp32) ----
__global__ void k_gemm1(const float* __restrict__ x, const float* __restrict__ W,
                        float* __restrict__ outm) {
  const int lane = threadIdx.x;          // one wave32 per block
  const int m0 = blockIdx.x * 16;
  const int n0 = blockIdx.y * 16;
  const int half = lane >> 4, l15 = lane & 15;
  v8f acc = {};
  const float* arow = x + (size_t)(m0 + l15) * KDIM + 2 * half;
#pragma unroll 4
  for (int kk = 0; kk < KDIM / 4; ++kk) {
    v2f a = *(const v2f*)(arow + 4 * kk);
    const int kbase = 4 * kk + 2 * half;
    v2f b;
    b.x = W[(size_t)(kbase + 0) * DCODE + n0 + l15];
    b.y = W[(size_t)(kbase + 1) * DCODE + n0 + l15];
    acc = __builtin_amdgcn_wmma_f32_16x16x4_f32(false, a, false, b,
                                                (short)0, acc, false, false);
  }
#pragma unroll
  for (int r = 0; r < 8; ++r)
    outm[(size_t)(m0 + r + 8 * half) * DCODE + n0 + l15] = acc[r];
}

// ---- Level-1 histogram of key[31:22] via 64KB LDS histogram, then flush ----
__global__ void k_hist1(const float* __restrict__ outm, const float* __restrict__ codes,
                        int* __restrict__ ghist) {
  __shared__ int lhist[16 * NB1];        // 64 KB LDS
  const int tid = threadIdx.x;
  for (int i = tid; i < 16 * NB1; i += blockDim.x) lhist[i] = 0;
  __syncthreads();

  const int lane = tid & 31, wave = tid >> 5;
  const int q0 = blockIdx.y * 16;
  const int half = lane >> 4, l15 = lane & 15;
  for (int nt = blockIdx.x * WPB + wave; nt < NTILES; nt += NBX * WPB) {
    const int n0 = nt * 16;
    const int nnext = nt + NBX * WPB;
    if (nnext < NTILES)
      __builtin_prefetch(codes + (size_t)(nnext * 16 + l15) * DCODE, 0, 0);
    v8f acc = dot_tile(outm, codes, q0, n0, lane);
#pragma unroll
    for (int r = 0; r < 8; ++r) {
      const int mi = r + 8 * half;                      // local query 0..15
      const unsigned bin = orderKey(acc[r]) >> 22;      // 10 bits
      atomicAdd(&lhist[mi * NB1 + bin], 1);
    }
  }
  __syncthreads();
  for (int i = tid; i < 16 * NB1; i += blockDim.x) {
    const int c = lhist[i];
    if (c) atomicAdd(&ghist[(size_t)(q0 + (i >> 10)) * HSTRIDE + (i & (NB1 - 1))], c);
  }
}

// ---- Refinement histograms (filtered, sparse -> global atomics) ----
__global__ void k_hist23(const float* __restrict__ outm, const float* __restrict__ codes,
                         const unsigned* __restrict__ prefix,
                         int matchShift, int binShift, int* __restrict__ ghist) {
  const int tid = threadIdx.x;
  const int lane = tid & 31, wave = tid >> 5;
  const int q0 = blockIdx.y * 16;
  const int half = lane >> 4, l15 = lane & 15;
  for (int nt = blockIdx.x * WPB + wave; nt < NTILES; nt += NBX * WPB) {
    const int n0 = nt * 16;
    v8f acc = dot_tile(outm, codes, q0, n0, lane);
#pragma unroll
    for (int r = 0; r < 8; ++r) {
      const int m = q0 + r + 8 * half;
      const unsigned key = orderKey(acc[r]);
      if ((key >> matchShift) == prefix[m]) {
        const int bin = (key >> binShift) & (NB23 - 1);
        atomicAdd(&ghist[(size_t)m * HSTRIDE + bin], 1);
      }
    }
  }
}

// ---- Radix-select step: scan bins descending, pick bin holding the want-th ----
__global__ void k_select(const int* __restrict__ ghist, int nbins, int levelBits,
                         const int* __restrict__ Kp, int level,
                         unsigned* __restrict__ prefix, int* __restrict__ selK) {
  const int m = blockIdx.x * blockDim.x + threadIdx.x;
  if (m >= NQ) return;
  const int want = (level == 1) ? Kp[0] : selK[m];
  const int* h = ghist + (size_t)m * HSTRIDE;
  int cum = 0, chosenBin = 0, newK = want;
  for (int bin = nbins - 1; bin >= 0; --bin) {
    const int c = h[bin];
    if (cum + c >= want) { chosenBin = bin; newK = want - cum; break; }
    cum += c;
  }
  const unsigned p = (level == 1) ? 0u : prefix[m];
  prefix[m] = (p << levelBits) | (unsigned)chosenBin;
  selK[m] = newK;
}

// ---- Final count pass (TDM double-buffered codes staging in LDS) ----
__global__ void k_count(const float* __restrict__ outm, const float* __restrict__ codes,
                        const int* __restrict__ labels,
                        const unsigned* __restrict__ kthKey,
                        int* __restrict__ counts,
                        unsigned* __restrict__ cand, int* __restrict__ candCount) {
  const int tid = threadIdx.x;
  const int lane = tid & 31;
  const int wave = __builtin_amdgcn_readfirstlane(tid >> 5);   // wave-uniform scalar
  const int q0 = blockIdx.y * 16;
  const int half = lane >> 4, l15 = lane & 15;

  const int ntFirst = blockIdx.x * WPBC + wave;
  const int stride  = NBX * WPBC;

#if HAVE_TDM
  __shared__ float stage[2 * WPBC * 16 * ROWP];   // 34 KB: 2 buffers x 4 waves
  const unsigned ldsBase = (unsigned)(uintptr_t)(void*)stage;  // LDS byte offset
  const unsigned bufBytes = 16u * ROWP * 4u;
  const unsigned myBase = ldsBase + (unsigned)wave * 2u * bufBytes;

  if (ntFirst < NTILES)
    tdm_load_tile(myBase, codes + (size_t)ntFirst * 16 * DCODE);

  int buf = 0;
  for (int nt = ntFirst; nt < NTILES; nt += stride) {
    const int ntNext = nt + stride;
    if (ntNext < NTILES) {
      tdm_load_tile(myBase + (unsigned)(buf ^ 1) * bufBytes,
                    codes + (size_t)ntNext * 16 * DCODE);
      __builtin_amdgcn_s_wait_tensorcnt((short)1);   // oldest (current buf) done
    } else {
      __builtin_amdgcn_s_wait_tensorcnt((short)0);
    }
    const float* brow = stage + (size_t)(wave * 2 + buf) * 16 * ROWP
                              + (size_t)l15 * ROWP + 2 * half;
    v8f acc = dot_tile_lds(outm, brow, q0, lane);
    const int n0 = nt * 16;
#else
  for (int nt = ntFirst; nt < NTILES; nt += stride) {
    const int n0 = nt * 16;
    v8f acc = dot_tile(outm, codes, q0, n0, lane);
#endif
    const int n = n0 + l15;
    const int lab = labels[n];
#pragma unroll
    for (int r = 0; r < 8; ++r) {
      const int m = q0 + r + 8 * half;
      const unsigned key = orderKey(acc[r]);
      const unsigned kth = kthKey[m];
      if (key > kth) {
        atomicAdd(&counts[m * NUM_CLASSES + lab], 1);
      } else if (key == kth) {
        const int idx = atomicAdd(candCount, 1);
        if (idx < MAX_CAND) cand[idx] = ((unsigned)m << 17) | (unsigned)n;
      }
    }
#if HAVE_TDM
    buf ^= 1;
#endif
  }
}

// ---- Resolve exact-key ties deterministically by smallest db index ----
__global__ void k_ties(const unsigned* __restrict__ cand, const int* __restrict__ candCount,
                       const int* __restrict__ labels, const int* __restrict__ selK,
                       int* __restrict__ counts) {
  const int m = blockIdx.x * blockDim.x + threadIdx.x;
  if (m >= NQ) return;
  const int total = min(candCount[0], MAX_CAND);
  const int rem = selK[m];
  for (int i = 0; i < total; ++i) {
    const unsigned e = cand[i];
    if ((int)(e >> 17) != m) continue;
    const int n = (int)(e & 0x1FFFFu);
    int rank = 0;
    for (int j = 0; j < total; ++j) {
      const unsigned f = cand[j];
      if ((int)(f >> 17) == m && (int)(f & 0x1FFFFu) < n) rank++;
    }
    if (rank < rem) counts[m * NUM_CLASSES + labels[n]] += 1;  // one thread per m
  }
}

// ---- probs = counts / K ----
__global__ void k_norm(const int* __restrict__ counts, const int* __restrict__ Kp,
                       float* __restrict__ probs) {
  const int i = blockIdx.x * blockDim.x + threadIdx.x;
  if (i < NQ * NUM_CLASSES) probs[i] = (float)counts[i] / (float)Kp[0];
}

extern "C" void kernel_launch(void* const* d_in, const int* in_sizes, int n_in,
                              void* d_out, int out_size, void* d_ws, size_t ws_size,
                              hipStream_t stream) {
  const float* x      = (const float*)d_in[0];
  const float* W      = (const float*)d_in[1];
  const float* codes  = (const float*)d_in[2];
  const int*   labels = (const int*)d_in[3];
  const int*   Kp     = (const int*)d_in[4];
  float* probs = (float*)d_out;

  char* ws = (char*)d_ws;
  size_t off = 0;
  float*    outm      = (float*)(ws + off);    off += (size_t)NQ * DCODE * 4;       // 64 KB
  int*      ghist     = (int*)(ws + off);      off += (size_t)NQ * HSTRIDE * 4;     // 2 MB
  unsigned* prefix    = (unsigned*)(ws + off); off += NQ * 4;
  int*      selK      = (int*)(ws + off);      off += NQ * 4;
  int*      counts    = (int*)(ws + off);      off += (size_t)NQ * NUM_CLASSES * 4; // 100 KB
  unsigned* cand      = (unsigned*)(ws + off); off += (size_t)MAX_CAND * 4;         // 64 KB
  int*      candCount = (int*)(ws + off);      off += 4;

  const dim3 gridTiles(NBX, NQ / 16);

  // out = x @ W
  k_gemm1<<<dim3(NQ / 16, DCODE / 16), 32, 0, stream>>>(x, W, outm);

  // Level 1: bits [31:22]
  hipMemsetAsync(ghist, 0, (size_t)NQ * HSTRIDE * 4, stream);
  k_hist1<<<gridTiles, 256, 0, stream>>>(outm, codes, ghist);
  k_select<<<8, 32, 0, stream>>>(ghist, NB1, 10, Kp, 1, prefix, selK);

  // Level 2: bits [21:11]
  hipMemsetAsync(ghist, 0, (size_t)NQ * HSTRIDE * 4, stream);
  k_hist23<<<gridTiles, 256, 0, stream>>>(outm, codes, prefix, 22, 11, ghist);
  k_select<<<8, 32, 0, stream>>>(ghist, NB23, 11, Kp, 2, prefix, selK);

  // Level 3: bits [10:0]  -> prefix becomes the exact 32-bit K-th key
  hipMemsetAsync(ghist, 0, (size_t)NQ * HSTRIDE * 4, stream);
  k_hist23<<<gridTiles, 256, 0, stream>>>(outm, codes, prefix, 11, 0, ghist);
  k_select<<<8, 32, 0, stream>>>(ghist, NB23, 11, Kp, 3, prefix, selK);

  // Count labels of the top-K set; collect exact-key ties
  hipMemsetAsync(counts, 0, (size_t)NQ * NUM_CLASSES * 4, stream);
  hipMemsetAsync(candCount, 0, 4, stream);
  k_count<<<gridTiles, WPBC * 32, 0, stream>>>(outm, codes, labels, prefix, counts,
                                               cand, candCount);
  k_ties<<<8, 32, 0, stream>>>(cand, candCount, labels, selK, counts);

  k_norm<<<(NQ * NUM_CLASSES + 255) / 256, 256, 0, stream>>>(counts, Kp, probs);
}